// GCN_56521769616159
// MI455X (gfx1250) — compile-verified
//
#include <hip/hip_runtime.h>
#include <hip/hip_bf16.h>

typedef __attribute__((ext_vector_type(2))) float v2f;
typedef __attribute__((ext_vector_type(8))) float v8f;

constexpr int N_NODES = 100000;
constexpr int E_EDGES = 1600000;
constexpr int IN_F = 128;
constexpr int H = 64;

// ---------------- degree / normalization ----------------

__global__ __launch_bounds__(256) void init_deg_kernel(float* deg, int n) {
    int i = blockIdx.x * 256 + threadIdx.x;
    if (i < n) deg[i] = 1.0f;  // self-loop contributes 1
}

__global__ __launch_bounds__(256) void deg_atomic_kernel(const int* __restrict__ dst,
                                                         float* deg, int e) {
    int i = blockIdx.x * 256 + threadIdx.x;
    if (i < e) atomicAdd(&deg[dst[i]], 1.0f);
}

__global__ __launch_bounds__(256) void dinv_kernel(float* deg, int n) {
    int i = blockIdx.x * 256 + threadIdx.x;
    if (i < n) deg[i] = rsqrtf(fmaxf(deg[i], 1.0f));
}

__global__ __launch_bounds__(256) void coef_kernel(const int* __restrict__ src,
                                                   const int* __restrict__ dst,
                                                   const float* __restrict__ dinv,
                                                   float* __restrict__ coef, int e) {
    int i = blockIdx.x * 256 + threadIdx.x;
    if (i < e) coef[i] = dinv[src[i]] * dinv[dst[i]];
}

// ---------------- fp32 WMMA GEMM: C[M x 64] = A[M x K] * W[K x 64] (+bias) ----------------
// 256 threads = 8 waves per block; block computes 32 rows x 64 cols.
// Wave w: row tile tm = (w>>2)*16, col tile tn = (w&3)*16.
// V_WMMA_F32_16X16X4_F32 operand layout (32-bit, wave32):
//   A 16x4:  lanes 0-15 -> M=lane, {v0,v1} = K{0,1}; lanes 16-31 -> M=lane-16, K{2,3}
//   B 4x16:  lanes 0-15 -> N=lane, {v0,v1} = K{0,1}; lanes 16-31 -> N=lane-16, K{2,3}
//   C/D 16x16: vgpr r, lane<16 -> (M=r,   N=lane); lane>=16 -> (M=r+8, N=lane-16)
template <int K, bool BIAS>
__global__ __launch_bounds__(256) void gemm_wmma_kernel(const float* __restrict__ A,
                                                        const float* __restrict__ W,
                                                        const float* __restrict__ bias,
                                                        float* __restrict__ C) {
    constexpr int LDA = K + 1;   // pad: K is a multiple of 64 -> avoid 64-bank conflicts
    constexpr int LDB = H + 1;   // 65
    __shared__ float sA[32 * LDA];
    __shared__ float sB[K * LDB];

    const int tid = threadIdx.x;
    const int m0  = blockIdx.x * 32;

    // Stage weights (K x 64) into LDS
    for (int i = tid; i < K * H; i += 256) {
        sB[(i / H) * LDB + (i % H)] = W[i];
    }
    // Stage A tile (32 x K) into LDS
    for (int i = tid; i < 32 * K; i += 256) {
        int r = i / K, c = i % K;
        sA[r * LDA + c] = A[(size_t)(m0 + r) * K + c];
    }
    __syncthreads();

    const int wave = tid >> 5;
    const int lane = tid & 31;
    const int half = lane >> 4;     // 0: K{0,1}/rows 0-7 ; 1: K{2,3}/rows 8-15
    const int lrow = lane & 15;
    const int tm = (wave >> 2) * 16;
    const int tn = (wave & 3) * 16;

    const float* aRow = &sA[(tm + lrow) * LDA];
    v8f acc = {};
#pragma unroll
    for (int k = 0; k < K; k += 4) {
        v2f a, b;
        a.x = aRow[k + 2 * half + 0];
        a.y = aRow[k + 2 * half + 1];
        b.x = sB[(k + 2 * half + 0) * LDB + tn + lrow];
        b.y = sB[(k + 2 * half + 1) * LDB + tn + lrow];
        acc = __builtin_amdgcn_wmma_f32_16x16x4_f32(
            /*neg_a=*/false, a, /*neg_b=*/false, b,
            /*c_mod=*/(short)0, acc, /*reuse_a=*/false, /*reuse_b=*/false);
    }

    const int col = tn + lrow;
    float bv = BIAS ? bias[col] : 0.0f;
#pragma unroll
    for (int r = 0; r < 8; ++r) {
        int row = m0 + tm + r + 8 * half;
        C[(size_t)row * H + col] = acc[r] + bv;
    }
}

// ---------------- residual + LayerNorm + ReLU (one wave32 per row, 2 feats/lane) ----------------

__global__ __launch_bounds__(256) void fuse_ln_relu_kernel(float* __restrict__ h,
                                                           const float* __restrict__ ori,
                                                           const float* __restrict__ lnw,
                                                           const float* __restrict__ lnb,
                                                           int n) {
    int wave = (blockIdx.x * 256 + threadIdx.x) >> 5;
    int lane = threadIdx.x & 31;
    if (wave >= n) return;
    size_t base = (size_t)wave * H;
    float x0 = h[base + lane]      + ori[base + lane];
    float x1 = h[base + lane + 32] + ori[base + lane + 32];

    float s = x0 + x1;
#pragma unroll
    for (int off = 16; off > 0; off >>= 1) s += __shfl_xor(s, off, 32);
    float mu = s * (1.0f / 64.0f);
    float d0 = x0 - mu, d1 = x1 - mu;
    float v = d0 * d0 + d1 * d1;
#pragma unroll
    for (int off = 16; off > 0; off >>= 1) v += __shfl_xor(v, off, 32);
    float rstd = rsqrtf(v * (1.0f / 64.0f) + 1e-5f);

    float y0 = d0 * rstd * lnw[lane]      + lnb[lane];
    float y1 = d1 * rstd * lnw[lane + 32] + lnb[lane + 32];
    h[base + lane]      = fmaxf(y0, 0.0f);
    h[base + lane + 32] = fmaxf(y1, 0.0f);
}

// ---------------- scatter: out = selfloop + bias, then edge atomics ----------------

__global__ __launch_bounds__(256) void scatter_init_kernel(float* __restrict__ out,
                                                           const float* __restrict__ tmp,
                                                           const float* __restrict__ dinv,
                                                           const float* __restrict__ bias,
                                                           int n) {
    int idx = blockIdx.x * 256 + threadIdx.x;  // over n*64
    if (idx >= n * H) return;
    int node = idx >> 6;
    int f = idx & 63;
    float di = dinv[node];
    out[idx] = tmp[idx] * di * di + bias[f];
}

__global__ __launch_bounds__(256) void scatter_edges_kernel(const int* __restrict__ src,
                                                            const int* __restrict__ dst,
                                                            const float* __restrict__ tmp,
                                                            const float* __restrict__ coef,
                                                            float* __restrict__ out,
                                                            int e) {
    int idx = blockIdx.x * 256 + threadIdx.x;  // over e*64
    int ed = idx >> 6;
    if (ed >= e) return;
    int f = idx & 63;
    int s = src[ed], d = dst[ed];
    float v = tmp[(size_t)s * H + f] * coef[ed];
    atomicAdd(&out[(size_t)d * H + f], v);
}

// ---------------- launcher ----------------

extern "C" void kernel_launch(void* const* d_in, const int* in_sizes, int n_in,
                              void* d_out, int out_size, void* d_ws, size_t ws_size,
                              hipStream_t stream) {
    const float* in_feat = (const float*)d_in[0];                 // [N, 128]
    const int*   src     = (const int*)d_in[1];                   // edge_index[0]
    const int*   dst     = ((const int*)d_in[1]) + E_EDGES;       // edge_index[1]
    const float* lin_w   = (const float*)d_in[2];                 // [128, 64]
    const float* lin_b   = (const float*)d_in[3];                 // [64]
    const float* conv_w  = (const float*)d_in[4];                 // [64, 64]
    const float* conv_b  = (const float*)d_in[5];                 // [64]
    const float* ln_w    = (const float*)d_in[6];                 // [3, 64]
    const float* ln_b    = (const float*)d_in[7];                 // [3, 64]
    float* out = (float*)d_out;                                   // [N, 64]

    float* ws   = (float*)d_ws;
    float* dinv = ws;  ws += N_NODES;                 // 0.4 MB
    float* coef = ws;  ws += E_EDGES;                 // 6.4 MB
    float* ori  = ws;  ws += (size_t)N_NODES * H;     // 25.6 MB
    float* h    = ws;  ws += (size_t)N_NODES * H;     // 25.6 MB
    float* tmp  = ws;  ws += (size_t)N_NODES * H;     // 25.6 MB

    const int nBlkN   = (N_NODES + 255) / 256;
    const int nBlkE   = (E_EDGES + 255) / 256;
    const int nBlkNH  = (N_NODES * H) / 256;          // 25000 exactly
    const int nBlkEH  = (E_EDGES * H + 255) / 256;    // 400000
    const int nBlkRow = N_NODES / 8;                  // 12500 (8 rows/block in fuse)
    const int nBlkGem = N_NODES / 32;                 // 3125 exactly

    // Normalization coefficients
    init_deg_kernel<<<nBlkN, 256, 0, stream>>>(dinv, N_NODES);
    deg_atomic_kernel<<<nBlkE, 256, 0, stream>>>(dst, dinv, E_EDGES);
    dinv_kernel<<<nBlkN, 256, 0, stream>>>(dinv, N_NODES);
    coef_kernel<<<nBlkE, 256, 0, stream>>>(src, dst, dinv, coef, E_EDGES);

    // ori = in_feat @ lin_w + lin_b
    gemm_wmma_kernel<IN_F, true><<<nBlkGem, 256, 0, stream>>>(in_feat, lin_w, lin_b, ori);

    for (int i = 0; i < 3; ++i) {
        const float* inp;
        if (i == 0) {
            inp = ori;
        } else {
            fuse_ln_relu_kernel<<<nBlkRow, 256, 0, stream>>>(h, ori, ln_w + i * H, ln_b + i * H,
                                                             N_NODES);
            inp = h;
        }
        gemm_wmma_kernel<H, false><<<nBlkGem, 256, 0, stream>>>(inp, conv_w, nullptr, tmp);
        float* target = (i == 2) ? out : h;
        scatter_init_kernel<<<nBlkNH, 256, 0, stream>>>(target, tmp, dinv, conv_b, N_NODES);
        scatter_edges_kernel<<<nBlkEH, 256, 0, stream>>>(src, dst, tmp, coef, target, E_EDGES);
    }
}